// GNNClassifier_51780125720665
// MI455X (gfx1250) — compile-verified
//
#include <hip/hip_runtime.h>
#include <hip/hip_bf16.h>
#include <math.h>

typedef float v2f __attribute__((ext_vector_type(2)));
typedef float v8f __attribute__((ext_vector_type(8)));

#define D128 128
#define HEADS 4
#define NETYPE 4
#define NSTEPS 3
#define GSEG 64
#define NCLS 10
#define BN_EPS 1e-5f

// register blocking: each wave computes (MT*16) x (NT*16) of C
#define MT 4
#define NT 2

// ---------------- float ordered-key helpers for atomic max on floats -------
__device__ __forceinline__ unsigned fenc(float f) {
    unsigned u = __float_as_uint(f);
    return (u & 0x80000000u) ? ~u : (u | 0x80000000u);
}
__device__ __forceinline__ float fdec(unsigned u) {
    u = (u & 0x80000000u) ? (u & 0x7fffffffu) : ~u;
    return __uint_as_float(u);
}

// ---------------- utility kernels ------------------------------------------
__global__ void zero_f32(float* __restrict__ p, size_t n) {
    size_t i = (size_t)blockIdx.x * blockDim.x + threadIdx.x;
    size_t stride = (size_t)gridDim.x * blockDim.x;
    for (; i < n; i += stride) p[i] = 0.0f;
}

__global__ void elu_copy(float* __restrict__ dst, const float* __restrict__ src, size_t n) {
    size_t i = (size_t)blockIdx.x * blockDim.x + threadIdx.x;
    if (i >= n) return;
    float x = src[i];
    dst[i] = x > 0.0f ? x : (expf(x) - 1.0f);
}

__global__ void elu_inplace(float* __restrict__ x, size_t n) {
    size_t i = (size_t)blockIdx.x * blockDim.x + threadIdx.x;
    if (i >= n) return;
    float v = x[i];
    x[i] = v > 0.0f ? v : (expf(v) - 1.0f);
}

__global__ void add_inplace(float* __restrict__ x, const float* __restrict__ y, size_t n) {
    size_t i = (size_t)blockIdx.x * blockDim.x + threadIdx.x;
    if (i >= n) return;
    x[i] += y[i];
}

// ---------------- generic fp32 WMMA GEMM: C[M,Nout] = A[M,K] @ W[Nout,K]^T + b
// Block = 8 waves covering a 128x128 C tile; each wave owns a 64x32 subtile
// (MT=4 m-tiles x NT=2 n-tiles) accumulated with V_WMMA_F32_16X16X4_F32.
// Per k-quad: 4 A-frag + 2 B-frag b64 loads feed 8 WMMAs (6 B/WMMA vs 16 B
// unblocked), with 8 independent accumulation chains for latency hiding.
// A-frag: lanes 0-15 = rows, K={k,k+1}; lanes 16-31 same rows K={k+2,k+3}.
// B-frag: identical pattern over the columns of C (rows of W).
// C: VGPR j -> row tile_m + j + 8*half, col tile_n + (lane&15).
__global__ void __launch_bounds__(256)
gemm_bias_wmma(const float* __restrict__ A, const float* __restrict__ W,
               const float* __restrict__ bias, float* __restrict__ C,
               int M, int K, int Nout, int act)
{
    int w    = threadIdx.x >> 5;
    int lane = threadIdx.x & 31;
    int half = lane >> 4;
    int r    = lane & 15;
    int wm   = w >> 2;            // 0..1
    int wn   = w & 3;             // 0..3

    int m0 = blockIdx.x * 128 + wm * 64;
    int n0 = blockIdx.y * 128 + wn * 32;
    if (m0 >= M || n0 >= Nout) return;

    // Clamp OOB lanes to row 0: their WMMA contributions land only in rows/cols
    // that are never stored, so garbage is harmless.
    const float* arow[MT];
    const float* brow[NT];
#pragma unroll
    for (int mt = 0; mt < MT; ++mt) {
        int mrow = m0 + mt * 16 + r;
        arow[mt] = A + (size_t)(mrow < M ? mrow : 0) * K + 2 * half;
    }
#pragma unroll
    for (int nt = 0; nt < NT; ++nt) {
        int nrow = n0 + nt * 16 + r;
        brow[nt] = W + (size_t)(nrow < Nout ? nrow : 0) * K + 2 * half;
    }

    v8f acc[MT][NT];
#pragma unroll
    for (int mt = 0; mt < MT; ++mt)
#pragma unroll
        for (int nt = 0; nt < NT; ++nt)
            acc[mt][nt] = (v8f){0.f,0.f,0.f,0.f,0.f,0.f,0.f,0.f};

#pragma unroll 4
    for (int k = 0; k < K; k += 4) {
        v2f af[MT], bf[NT];
#pragma unroll
        for (int mt = 0; mt < MT; ++mt) af[mt] = *(const v2f*)(arow[mt] + k);
#pragma unroll
        for (int nt = 0; nt < NT; ++nt) bf[nt] = *(const v2f*)(brow[nt] + k);
#pragma unroll
        for (int mt = 0; mt < MT; ++mt)
#pragma unroll
            for (int nt = 0; nt < NT; ++nt)
                acc[mt][nt] = __builtin_amdgcn_wmma_f32_16x16x4_f32(
                    false, af[mt], false, bf[nt], (short)0, acc[mt][nt], false, false);
    }

#pragma unroll
    for (int nt = 0; nt < NT; ++nt) {
        int ncol = n0 + nt * 16 + r;
        bool nok = ncol < Nout;
        float bb = nok ? bias[ncol] : 0.0f;
#pragma unroll
        for (int mt = 0; mt < MT; ++mt) {
#pragma unroll
            for (int j = 0; j < 8; ++j) {
                int m = m0 + mt * 16 + half * 8 + j;
                if (nok && m < M) {
                    float v = acc[mt][nt][j] + bb;
                    if (act) v = fmaxf(v, 0.0f);
                    C[(size_t)m * Nout + ncol] = v;
                }
            }
        }
    }
}

// ---------------- GatedGraphConv: gather msg + scatter-add ------------------
__global__ void ggc_scatter(const float* __restrict__ hall, const int* __restrict__ src,
                            const int* __restrict__ dst, const int* __restrict__ et,
                            float* __restrict__ a, int E, int N)
{
    int e = blockIdx.x * 2 + (threadIdx.x >> 7);
    if (e >= E) return;
    int d = threadIdx.x & 127;
    int t = et[e], s = src[e], dd = dst[e];
    float v = hall[((size_t)t * N + s) * D128 + d];
    atomicAdd(&a[(size_t)dd * D128 + d], v);
}

// ---------------- GRU cell update ------------------------------------------
__global__ void gru_update(float* __restrict__ hv, const float* __restrict__ gi,
                           const float* __restrict__ gh, size_t total)
{
    size_t idx = (size_t)blockIdx.x * blockDim.x + threadIdx.x;
    if (idx >= total) return;
    size_t n = idx / D128;
    int dcol = (int)(idx - n * D128);
    size_t b = n * (3 * D128) + dcol;
    float ir = gi[b], iz = gi[b + D128], inn = gi[b + 2 * D128];
    float hr = gh[b], hz = gh[b + D128], hn  = gh[b + 2 * D128];
    float rg = 1.0f / (1.0f + expf(-(ir + hr)));
    float zg = 1.0f / (1.0f + expf(-(iz + hz)));
    float ng = tanhf(inn + rg * hn);
    hv[idx] = (1.0f - zg) * ng + zg * hv[idx];
}

// ---------------- BatchNorm: coalesced column partial sums + apply ----------
__global__ void bn_partial(const float* __restrict__ x, float* __restrict__ sums,
                           float* __restrict__ sqs, int Nrows, int C)
{
    int c = blockIdx.x * 64 + threadIdx.x;        // block = (64,4)
    int rstart = blockIdx.y * 4 + threadIdx.y;
    int rstride = gridDim.y * 4;
    float s = 0.0f, q = 0.0f;
    for (int row = rstart; row < Nrows; row += rstride) {
        float v = x[(size_t)row * C + c];
        s += v; q += v * v;
    }
    __shared__ float ls[4][64], lq[4][64];
    ls[threadIdx.y][threadIdx.x] = s;
    lq[threadIdx.y][threadIdx.x] = q;
    __syncthreads();
    if (threadIdx.y == 0) {
        s = ls[0][threadIdx.x] + ls[1][threadIdx.x] + ls[2][threadIdx.x] + ls[3][threadIdx.x];
        q = lq[0][threadIdx.x] + lq[1][threadIdx.x] + lq[2][threadIdx.x] + lq[3][threadIdx.x];
        atomicAdd(&sums[c], s);
        atomicAdd(&sqs[c], q);
    }
}

__global__ void bn_apply(float* __restrict__ x, const float* __restrict__ sums,
                         const float* __restrict__ sqs, const float* __restrict__ g,
                         const float* __restrict__ b, int Nrows, int C)
{
    size_t idx = (size_t)blockIdx.x * blockDim.x + threadIdx.x;
    size_t tot = (size_t)Nrows * C;
    if (idx >= tot) return;
    int c = (int)(idx % C);
    float inv = 1.0f / (float)Nrows;
    float mu = sums[c] * inv;
    float var = sqs[c] * inv - mu * mu;
    x[idx] = (x[idx] - mu) * rsqrtf(var + BN_EPS) * g[c] + b[c];
}

// ---------------- GATv2 edge kernels ---------------------------------------
__global__ void gat_logits(const float* __restrict__ fs, const float* __restrict__ fd,
                           const float* __restrict__ attnW, const int* __restrict__ src,
                           const int* __restrict__ dst, float* __restrict__ zbuf,
                           unsigned* __restrict__ mkey, int E)
{
    int e = blockIdx.x;
    int h = threadIdx.x >> 5, lane = threadIdx.x & 31;
    int s = src[e], d = dst[e];
    const float4 a4 = *(const float4*)(fs + (size_t)s * 512 + h * D128 + lane * 4);
    const float4 b4 = *(const float4*)(fd + (size_t)d * 512 + h * D128 + lane * 4);
    const float4 w4 = *(const float4*)(attnW + h * D128 + lane * 4);
    float p = 0.0f, v;
    v = a4.x + b4.x; v = v > 0.0f ? v : 0.2f * v; p += v * w4.x;
    v = a4.y + b4.y; v = v > 0.0f ? v : 0.2f * v; p += v * w4.y;
    v = a4.z + b4.z; v = v > 0.0f ? v : 0.2f * v; p += v * w4.z;
    v = a4.w + b4.w; v = v > 0.0f ? v : 0.2f * v; p += v * w4.w;
    for (int o = 16; o > 0; o >>= 1) p += __shfl_down(p, o);
    if (lane == 0) {
        zbuf[(size_t)e * HEADS + h] = p;
        atomicMax(&mkey[(size_t)d * HEADS + h], fenc(p));
    }
}

__global__ void gat_exp(float* __restrict__ zbuf, const int* __restrict__ dst,
                        const unsigned* __restrict__ mkey, float* __restrict__ ssum, int E)
{
    int idx = blockIdx.x * blockDim.x + threadIdx.x;
    if (idx >= E * HEADS) return;
    int e = idx >> 2, h = idx & 3;
    int d = dst[e];
    float z = expf(zbuf[idx] - fdec(mkey[(size_t)d * HEADS + h]));
    zbuf[idx] = z;
    atomicAdd(&ssum[(size_t)d * HEADS + h], z);
}

__global__ void gat_agg(const float* __restrict__ zbuf, const float* __restrict__ ssum,
                        const float* __restrict__ fs, const int* __restrict__ src,
                        const int* __restrict__ dst, float* __restrict__ h1, int E)
{
    int e = blockIdx.x;
    int s = src[e], d = dst[e];
    __shared__ float sa[HEADS];
    if (threadIdx.x < HEADS)
        sa[threadIdx.x] = zbuf[(size_t)e * HEADS + threadIdx.x] /
                          ssum[(size_t)d * HEADS + threadIdx.x];
    __syncthreads();
    const float* pf = fs + (size_t)s * 512;
    float* ph = h1 + (size_t)d * 512;
    for (int i = threadIdx.x; i < 512; i += 256)
        atomicAdd(&ph[i], sa[i >> 7] * pf[i]);
}

// ---------------- Global attention pooling ---------------------------------
__global__ void gate_dot(const float* __restrict__ hr, const float* __restrict__ gw,
                         const float* __restrict__ gb, const int* __restrict__ gid,
                         float* __restrict__ gateval, unsigned* __restrict__ gkey, int N)
{
    int wave = threadIdx.x >> 5, lane = threadIdx.x & 31;
    int n = blockIdx.x * 8 + wave;
    if (n >= N) return;
    const float4 h4 = *(const float4*)(hr + (size_t)n * D128 + lane * 4);
    const float4 w4 = *(const float4*)(gw + lane * 4);
    float p = h4.x * w4.x + h4.y * w4.y + h4.z * w4.z + h4.w * w4.w;
    for (int o = 16; o > 0; o >>= 1) p += __shfl_down(p, o);
    if (lane == 0) {
        float g = p + gb[0];
        gateval[n] = g;
        atomicMax(&gkey[gid[n]], fenc(g));
    }
}

__global__ void gate_exp(const float* __restrict__ gateval, const int* __restrict__ gid,
                         const unsigned* __restrict__ gkey, float* __restrict__ gsum,
                         float* __restrict__ gatez, int N)
{
    int n = blockIdx.x * blockDim.x + threadIdx.x;
    if (n >= N) return;
    int g = gid[n];
    float z = expf(gateval[n] - fdec(gkey[g]));
    gatez[n] = z;
    atomicAdd(&gsum[g], z);
}

__global__ void pool_kernel(const float* __restrict__ hr, const float* __restrict__ gatez,
                            const float* __restrict__ gsum, const int* __restrict__ gid,
                            float* __restrict__ hg, int N)
{
    int n = blockIdx.x;
    int g = gid[n];
    float ga = gatez[n] / gsum[g];
    int d = threadIdx.x;
    atomicAdd(&hg[g * D128 + d], ga * hr[(size_t)n * D128 + d]);
}

// ---------------- host-side helpers ----------------------------------------
static inline void launch_gemm(const float* A, const float* W, const float* b, float* C,
                               int M, int K, int Nout, int act, hipStream_t s)
{
    dim3 grid((M + 127) / 128, (Nout + 127) / 128);
    gemm_bias_wmma<<<grid, dim3(256), 0, s>>>(A, W, b, C, M, K, Nout, act);
}

static inline void launch_zero(float* p, size_t n, hipStream_t s)
{
    int blocks = (int)((n + 256 * 8 - 1) / (256 * 8));
    if (blocks < 1) blocks = 1;
    zero_f32<<<blocks, 256, 0, s>>>(p, n);
}

extern "C" void kernel_launch(void* const* d_in, const int* in_sizes, int n_in,
                              void* d_out, int out_size, void* d_ws, size_t ws_size,
                              hipStream_t stream)
{
    // ---- inputs ----
    const float* feat     = (const float*)d_in[0];
    const int*   src      = (const int*)d_in[1];
    const int*   dst      = (const int*)d_in[2];
    const int*   etype    = (const int*)d_in[3];
    const int*   gid      = (const int*)d_in[4];
    const float* ggc_W    = (const float*)d_in[5];
    const float* ggc_b    = (const float*)d_in[6];
    const float* gru_Wih  = (const float*)d_in[7];
    const float* gru_Whh  = (const float*)d_in[8];
    const float* gru_bih  = (const float*)d_in[9];
    const float* gru_bhh  = (const float*)d_in[10];
    const float* gat_Wsrc = (const float*)d_in[11];
    const float* gat_bsrc = (const float*)d_in[12];
    const float* gat_Wdst = (const float*)d_in[13];
    const float* gat_bdst = (const float*)d_in[14];
    const float* gat_attn = (const float*)d_in[15];
    const float* skip_W   = (const float*)d_in[16];
    const float* skip_b   = (const float*)d_in[17];
    const float* bn1_g    = (const float*)d_in[18];
    const float* bn1_b    = (const float*)d_in[19];
    const float* bn2_g    = (const float*)d_in[20];
    const float* bn2_b    = (const float*)d_in[21];
    const float* red_W    = (const float*)d_in[22];
    const float* red_b    = (const float*)d_in[23];
    const float* gate_W   = (const float*)d_in[24];
    const float* gate_b   = (const float*)d_in[25];
    const float* cls_W1   = (const float*)d_in[26];
    const float* cls_b1   = (const float*)d_in[27];
    const float* cls_W2   = (const float*)d_in[28];
    const float* cls_b2   = (const float*)d_in[29];

    const int N = in_sizes[0] / D128;
    const int E = in_sizes[1];
    const size_t F = (size_t)N * D128;

    // ---- workspace layout (floats); total ~16F + small ~415 MB ----
    float* ws   = (float*)d_ws;
    float* hv   = ws;                  // F   : GRU state; later hr
    float* axb  = ws + F;              // F   : aggregated msgs; later h2
    float* gi   = ws + 2 * F;          // 3F  : GRU input gates; later h1 (4F: 2F..6F)
    float* gh   = ws + 5 * F;          // 3F
    float* hall = ws + 8 * F;          // 4F  : per-etype msgs; later fs; later skip
    float* fdb  = ws + 12 * F;         // 4F  : fd
    float* small = ws + 16 * F;
    float*    zbuf    = small;                              // E*4
    float*    ssum    = zbuf + (size_t)E * HEADS;           // N*4
    unsigned* mkey    = (unsigned*)(ssum + (size_t)N * HEADS); // N*4
    float*    sums    = (float*)(mkey + (size_t)N * HEADS); // 512
    float*    sqs     = sums + 512;                         // 512
    float*    gateval = sqs + 512;                          // N
    float*    gatez   = gateval + N;                        // N
    unsigned* gkey    = (unsigned*)(gatez + N);             // 64
    float*    gsum    = (float*)(gkey + GSEG);              // 64
    float*    hg      = gsum + GSEG;                        // 64*128
    float*    t1      = hg + GSEG * D128;                   // 64*64

    const size_t ND = F;
    const int eltBlocks512 = (int)(((size_t)N * 512 + 255) / 256);
    const int eltBlocks128 = (int)((ND + 255) / 256);

    // hv = feat
    hipMemcpyAsync(hv, feat, F * sizeof(float), hipMemcpyDeviceToDevice, stream);

    // ---- GatedGraphConv: NSTEPS iterations ----
    for (int step = 0; step < NSTEPS; ++step) {
        for (int t = 0; t < NETYPE; ++t)
            launch_gemm(hv, ggc_W + (size_t)t * D128 * D128, ggc_b + t * D128,
                        hall + (size_t)t * F, N, D128, D128, 0, stream);
        launch_zero(axb, ND, stream);
        ggc_scatter<<<(E + 1) / 2, 256, 0, stream>>>(hall, src, dst, etype, axb, E, N);
        launch_gemm(axb, gru_Wih, gru_bih, gi, N, D128, 3 * D128, 0, stream);
        launch_gemm(hv,  gru_Whh, gru_bhh, gh, N, D128, 3 * D128, 0, stream);
        gru_update<<<eltBlocks128, 256, 0, stream>>>(hv, gi, gh, ND);
    }

    // ---- h2 = BN(elu(hv)) ----
    elu_copy<<<eltBlocks128, 256, 0, stream>>>(axb, hv, ND);
    launch_zero(sums, 1024, stream);
    bn_partial<<<dim3(D128 / 64, 64), dim3(64, 4), 0, stream>>>(axb, sums, sqs, N, D128);
    bn_apply<<<eltBlocks128, 256, 0, stream>>>(axb, sums, sqs, bn2_g, bn2_b, N, D128);

    // ---- GATv2 ----
    float* fs = hall;    // reuse
    launch_gemm(axb, gat_Wsrc, gat_bsrc, fs,  N, D128, 512, 0, stream);
    launch_gemm(axb, gat_Wdst, gat_bdst, fdb, N, D128, 512, 0, stream);

    launch_zero((float*)mkey, (size_t)N * HEADS, stream);   // 0u == key floor (< enc(-inf))
    gat_logits<<<E, 128, 0, stream>>>(fs, fdb, gat_attn, src, dst, zbuf, mkey, E);
    launch_zero(ssum, (size_t)N * HEADS, stream);
    gat_exp<<<(E * HEADS + 255) / 256, 256, 0, stream>>>(zbuf, dst, mkey, ssum, E);

    float* h1 = gi;      // reuse 4F at [2F..6F]
    launch_zero(h1, (size_t)N * 512, stream);
    gat_agg<<<E, 256, 0, stream>>>(zbuf, ssum, fs, src, dst, h1, E);
    elu_inplace<<<eltBlocks512, 256, 0, stream>>>(h1, (size_t)N * 512);

    float* skip = hall;  // fs no longer needed after gat_agg
    launch_gemm(feat, skip_W, skip_b, skip, N, D128, 512, 0, stream);
    add_inplace<<<eltBlocks512, 256, 0, stream>>>(h1, skip, (size_t)N * 512);

    launch_zero(sums, 1024, stream);
    bn_partial<<<dim3(512 / 64, 64), dim3(64, 4), 0, stream>>>(h1, sums, sqs, N, 512);
    bn_apply<<<eltBlocks512, 256, 0, stream>>>(h1, sums, sqs, bn1_g, bn1_b, N, 512);

    float* hr = hv;      // reuse
    launch_gemm(h1, red_W, red_b, hr, N, 512, D128, 0, stream);

    // ---- Global attention pooling ----
    launch_zero((float*)gkey, GSEG, stream);
    gate_dot<<<(N + 7) / 8, 256, 0, stream>>>(hr, gate_W, gate_b, gid, gateval, gkey, N);
    launch_zero(gsum, GSEG, stream);
    gate_exp<<<(N + 255) / 256, 256, 0, stream>>>(gateval, gid, gkey, gsum, gatez, N);
    launch_zero(hg, (size_t)GSEG * D128, stream);
    pool_kernel<<<N, D128, 0, stream>>>(hr, gatez, gsum, gid, hg, N);

    // ---- classifier ----
    launch_gemm(hg, cls_W1, cls_b1, t1, GSEG, D128, 64, 1, stream);
    launch_gemm(t1, cls_W2, cls_b2, (float*)d_out, GSEG, 64, NCLS, 0, stream);
}